// Model_33088428048398
// MI455X (gfx1250) — compile-verified
//
#include <hip/hip_runtime.h>
#include <hip/hip_bf16.h>

// ---------------------------------------------------------------------------
// 2-layer GCN for gfx1250 (MI455X).
//   h1 = relu( Ddst^-1/2 * A^T * (Dsrc^-1/2 * X) W1 + b1 )   [100000 x 64]
//   h2 =       Ddst^-1/2 * A^T * (Dsrc^-1/2 * h1) W2 + b2    [100000 x 40]
// GEMMs use V_WMMA_F32_16X16X4_F32 (fp32 kept: GEMMs are ~2 GFLOP total,
// the kernel is bound by the 1.6M-edge gather/scatter, ~1.3 GB @ 23.3 TB/s).
// Edge scatter uses coalesced float4 gathers + global_atomic_add_f32 into an
// accumulator that lives in the 192 MB L2.
// ---------------------------------------------------------------------------

typedef __attribute__((ext_vector_type(2))) float v2f;
typedef __attribute__((ext_vector_type(8))) float v8f;

#define IN_DIM  128
#define HID_DIM 64
#define OUT_DIM 40

// ---------------- degree / norm kernels ------------------------------------

__global__ void k_degrees(const int* __restrict__ src, const int* __restrict__ dst,
                          float* __restrict__ deg_out, float* __restrict__ deg_in,
                          int n_edges) {
    int e = blockIdx.x * blockDim.x + threadIdx.x;
    if (e < n_edges) {
        atomicAdd(&deg_out[src[e]], 1.0f);
        atomicAdd(&deg_in[dst[e]], 1.0f);
    }
}

__global__ void k_make_norms(float* __restrict__ ns, float* __restrict__ nd, int n) {
    int i = blockIdx.x * blockDim.x + threadIdx.x;
    if (i < n) {
        float a = ns[i]; a = (a < 1.0f) ? 1.0f : a; ns[i] = 1.0f / sqrtf(a);
        float b = nd[i]; b = (b < 1.0f) ? 1.0f : b; nd[i] = 1.0f / sqrtf(b);
    }
}

// ---------------- WMMA GEMM: C[n_rows x ldc] = (A * norm) @ W --------------
// One wave -> one 16x16 fp32 C tile, K accumulated 4 at a time with
// v_wmma_f32_16x16x4_f32.  blockIdx.x = M tile, wave id = N tile.
//
// Fragment layouts (ISA 7.12.2, 32-bit):
//   A 16x4 : lane<16 -> M=lane,   holds K = {k0, k0+1} in v[0],v[1]
//            lane>=16-> M=lane-16,holds K = {k0+2,k0+3}
//   B 4x16 : lane<16 -> N=lane,   holds K = {k0, k0+1}
//            lane>=16-> N=lane-16,holds K = {k0+2,k0+3}
//   C 16x16: VGPR r, lane<16 -> (M=r,   N=lane); lane>=16 -> (M=r+8, N=lane-16)

template <int K>
__global__ void k_gemm_norm_wmma(const float* __restrict__ A, int lda,
                                 const float* __restrict__ norm,
                                 const float* __restrict__ W, int ldw, int ncols,
                                 float* __restrict__ C, int ldc, int n_rows) {
    const int lane   = threadIdx.x & 31;
    const int wave   = threadIdx.x >> 5;
    const int lane16 = lane & 15;
    const int half   = lane >> 4;

    const int m0 = blockIdx.x * 16;
    const int n0 = wave * 16;

    // A-side row for this lane (clamped so EXEC stays all-ones; tail rows
    // are masked only at the store).
    int mrow = m0 + lane16;
    if (mrow > n_rows - 1) mrow = n_rows - 1;
    const float nrm = norm[mrow];

    // B-side column for this lane; columns beyond ncols contribute zeros.
    const int   nB    = n0 + lane16;
    const int   nBc   = (nB < ncols) ? nB : (ncols - 1);
    const float bmask = (nB < ncols) ? 1.0f : 0.0f;

    const float* Arow = A + (long long)mrow * lda + half * 2;
    const float* Wcol = W + (long long)(half * 2) * ldw + nBc;

    v8f acc = {};
#pragma unroll
    for (int k0 = 0; k0 < K; k0 += 4) {
        v2f a = *(const v2f*)(Arow + k0);     // K even, lda even -> 8B aligned
        a *= nrm;                              // fold Dsrc^-1/2 into A
        v2f b;
        b.x = Wcol[(long long)k0 * ldw] * bmask;
        b.y = Wcol[(long long)(k0 + 1) * ldw] * bmask;
        acc = __builtin_amdgcn_wmma_f32_16x16x4_f32(
            /*neg_a=*/false, a, /*neg_b=*/false, b,
            /*c_mod=*/(short)0, acc, /*reuse_a=*/false, /*reuse_b=*/false);
    }

#pragma unroll
    for (int r = 0; r < 8; ++r) {
        const int row = m0 + half * 8 + r;
        if (row < n_rows) C[(long long)row * ldc + n0 + lane16] = acc[r];
    }
}

// ---------------- edge scatter-add -----------------------------------------
// thread t -> edge e = t / nchunks, chunk c = t % nchunks (4 floats / chunk).
// Consecutive lanes read consecutive 16B of one source row -> coalesced.

__global__ void k_scatter_add(const float* __restrict__ hw, int ldh,
                              const int* __restrict__ src, const int* __restrict__ dst,
                              float* __restrict__ m, int ldm,
                              int nchunks, int n_edges) {
    long long t = (long long)blockIdx.x * blockDim.x + threadIdx.x;
    int e = (int)(t / nchunks);
    int c = (int)(t % nchunks);
    if (e >= n_edges) return;
    const int s = src[e];
    const int d = dst[e];
    const float4 v = *(const float4*)(hw + (long long)s * ldh + c * 4);
    float* p = m + (long long)d * ldm + c * 4;
    atomicAdd(p + 0, v.x);
    atomicAdd(p + 1, v.y);
    atomicAdd(p + 2, v.z);
    atomicAdd(p + 3, v.w);
}

// ---------------- epilogue: out = m * norm_dst + bias (opt relu) -----------

__global__ void k_finalize(const float* __restrict__ m,
                           const float* __restrict__ norm_dst,
                           const float* __restrict__ bias,
                           float* __restrict__ out,
                           int ncols, long long total, int do_relu) {
    long long i = (long long)blockIdx.x * blockDim.x + threadIdx.x;
    if (i < total) {
        int n = (int)(i / ncols);
        int j = (int)(i % ncols);
        float v = m[i] * norm_dst[n] + bias[j];
        if (do_relu) v = (v > 0.0f) ? v : 0.0f;
        out[i] = v;
    }
}

// ---------------------------------------------------------------------------

extern "C" void kernel_launch(void* const* d_in, const int* in_sizes, int n_in,
                              void* d_out, int out_size, void* d_ws, size_t ws_size,
                              hipStream_t stream) {
    const float* feats = (const float*)d_in[0];
    const float* W1    = (const float*)d_in[1];
    const float* b1    = (const float*)d_in[2];
    const float* W2    = (const float*)d_in[3];
    const float* b2    = (const float*)d_in[4];
    const int*   src   = (const int*)d_in[5];
    const int*   dst   = (const int*)d_in[6];
    float*       out   = (float*)d_out;

    const int n_nodes = in_sizes[0] / IN_DIM;   // 100000
    const int n_edges = in_sizes[5];            // 1600000
    const int mtiles  = (n_nodes + 15) / 16;    // 6250 (exact)

    // ---- workspace layout (floats), 256-float aligned regions -------------
    const long long NPAD = ((long long)n_nodes + 255) & ~255LL;
    float* norm_src = (float*)d_ws;             // [NPAD]
    float* norm_dst = norm_src + NPAD;          // [NPAD]
    float* hw       = norm_dst + NPAD;          // [n_nodes * 64] (layer2: *48)
    float* m        = hw + (long long)n_nodes * HID_DIM; // [n_nodes * 64]

    const int BLK = 256;

    // ---- degrees & norms (shared by both layers) --------------------------
    hipMemsetAsync(norm_src, 0, 2 * NPAD * sizeof(float), stream);
    k_degrees<<<(n_edges + BLK - 1) / BLK, BLK, 0, stream>>>(src, dst, norm_src, norm_dst, n_edges);
    k_make_norms<<<(n_nodes + BLK - 1) / BLK, BLK, 0, stream>>>(norm_src, norm_dst, n_nodes);

    // ---- layer 1: hw = (X * nsrc) @ W1 ; m = scatter ; h1 = relu(...) -----
    k_gemm_norm_wmma<IN_DIM><<<mtiles, 4 * 32, 0, stream>>>(
        feats, IN_DIM, norm_src, W1, HID_DIM, HID_DIM, hw, HID_DIM, n_nodes);

    hipMemsetAsync(m, 0, (long long)n_nodes * HID_DIM * sizeof(float), stream);
    {
        long long tot = (long long)n_edges * (HID_DIM / 4);
        k_scatter_add<<<(unsigned)((tot + BLK - 1) / BLK), BLK, 0, stream>>>(
            hw, HID_DIM, src, dst, m, HID_DIM, HID_DIM / 4, n_edges);
    }
    {
        long long tot = (long long)n_nodes * HID_DIM;
        k_finalize<<<(unsigned)((tot + BLK - 1) / BLK), BLK, 0, stream>>>(
            m, norm_dst, b1, out, HID_DIM, tot, /*relu=*/1);   // h1 -> d_out[0:6.4M)
    }

    // ---- layer 2: hw = (h1 * nsrc) @ W2 (N padded 40->48) -----------------
    const int LDC2 = 48;  // 3 n-tiles of 16
    k_gemm_norm_wmma<HID_DIM><<<mtiles, 3 * 32, 0, stream>>>(
        out, HID_DIM, norm_src, W2, OUT_DIM, OUT_DIM, hw, LDC2, n_nodes);

    hipMemsetAsync(m, 0, (long long)n_nodes * OUT_DIM * sizeof(float), stream);
    {
        long long tot = (long long)n_edges * (OUT_DIM / 4);
        k_scatter_add<<<(unsigned)((tot + BLK - 1) / BLK), BLK, 0, stream>>>(
            hw, LDC2, src, dst, m, OUT_DIM, OUT_DIM / 4, n_edges);
    }
    {
        long long tot = (long long)n_nodes * OUT_DIM;
        float* out2 = out + (long long)n_nodes * HID_DIM;
        k_finalize<<<(unsigned)((tot + BLK - 1) / BLK), BLK, 0, stream>>>(
            m, norm_dst, b2, out2, OUT_DIM, tot, /*relu=*/0);  // h2
    }
}